// BoundaryFocalLoss_72773925864076
// MI455X (gfx1250) — compile-verified
//
#include <hip/hip_runtime.h>

// Boundary focal loss: streaming elementwise pass + scalar reduction.
// Memory-bound: 12 B/elem * 25.6M = ~307 MB -> ~13 us at 23.3 TB/s.
// Math path trimmed to 3 native transcendentals/elem (v_exp/v_log/v_rcp)
// so VALU stays under the bandwidth roof. WMMA (f32 16x16x4, ones-matrix
// trick) performs the exact cross-lane wave reduction in f32.

typedef __attribute__((ext_vector_type(2))) float v2f;
typedef __attribute__((ext_vector_type(4))) float v4f;
typedef __attribute__((ext_vector_type(8))) float v8f;
typedef __attribute__((ext_vector_type(4))) int   v4i;

#define T_LEN 200000           // row length from the reference
#define N_BLOCKS 4096
#define BLOCK_SZ 256

// Cross-lane sum of one float per lane using V_WMMA_F32_16X16X4_F32.
// A (16x4 f32, 2 VGPRs): lane m<16 -> A[m][0]=v, lane m>=16 -> A[m-16][2]=v,
// second VGPR = 0.  B = all ones (4x16).  D[m][n] = v[m] + v[m+16] =: s[m].
// C/D layout: vgpr r, lanes 0-15 hold row M=r, lanes 16-31 hold row M=r+8.
// lanes<16 see s[0..7] in c[0..7], lanes>=16 see s[8..15]; one xor-16
// shuffle finishes the 32-lane sum (result broadcast to every lane).
__device__ __forceinline__ float wave_sum_wmma(float v) {
    v2f a; a[0] = v;    a[1] = 0.0f;
    v2f b; b[0] = 1.0f; b[1] = 1.0f;
    v8f c = {};
    c = __builtin_amdgcn_wmma_f32_16x16x4_f32(false, a, false, b,
                                              (short)0, c, false, false);
    float local = ((c[0] + c[1]) + (c[2] + c[3])) +
                  ((c[4] + c[5]) + (c[6] + c[7]));
    return local + __shfl_xor(local, 16, 32);
}

__global__ void __launch_bounds__(BLOCK_SZ)
bfl_main(const float* __restrict__ x, const int* __restrict__ tg,
         const float* __restrict__ mk, float* __restrict__ partials,
         int nvec) {
    float accL = 0.0f, accM = 0.0f;
    const int stride = gridDim.x * blockDim.x;

    for (int g = blockIdx.x * blockDim.x + threadIdx.x; g < nvec; g += stride) {
        const int idx0 = g << 2;             // first element of this vec4
        const int row  = idx0 / T_LEN;
        const int c0   = idx0 - row * T_LEN; // column, multiple of 4
        const long rowOff = (long)row * T_LEN;

        // inputs/mask are stream-once: non-temporal loads keep cache for
        // the targets halo reuse between neighboring threads.
        const v4f xv = __builtin_nontemporal_load(
                           reinterpret_cast<const v4f*>(x) + g);
        const v4f mv = __builtin_nontemporal_load(
                           reinterpret_cast<const v4f*>(mk) + g);
        const v4i tm = reinterpret_cast<const v4i*>(tg)[g];

        // Halo int4s; replicate border value at row edges so the
        // corresponding transitions become 0 (matches trans[0]=0 and the
        // init-value-0 window padding of reduce_window).
        v4i tl, tr;
        if (c0 >= 4)
            tl = *reinterpret_cast<const v4i*>(tg + rowOff + c0 - 4);
        else { tl[0] = tl[1] = tl[2] = tl[3] = tm[0]; }
        if (c0 <= T_LEN - 8)
            tr = *reinterpret_cast<const v4i*>(tg + rowOff + c0 + 4);
        else { tr[0] = tr[1] = tr[2] = tr[3] = tm[3]; }

        const int tw[12] = {tl[0], tl[1], tl[2], tl[3],
                            tm[0], tm[1], tm[2], tm[3],
                            tr[0], tr[1], tr[2], tr[3]};
        int df[12];                           // df[k] = trans(c0-4+k)
        #pragma unroll
        for (int k = 1; k < 12; ++k) df[k] = (tw[k] != tw[k - 1]) ? 1 : 0;

        #pragma unroll
        for (int e = 0; e < 4; ++e) {
            int bnd = 0;                      // window-max of transitions
            #pragma unroll
            for (int k = e + 1; k <= e + 7; ++k) bnd |= df[k];

            const float xi  = xv[e];
            const float mi  = mv[e];
            const float pos = (tm[e] > 0) ? 1.0f : 0.0f;

            const float s  = fmaf(pos, 0.95f, 0.025f);       // label smoothing
            float weight   = fmaf((float)bnd, 4.0f, 1.0f);   // 1 + b*(5-1)

            const float mx = fmaxf(xi, 0.0f);
            const float en = __expf(-fabsf(xi));             // e = exp(-|x|), v_exp
            const float r  = __builtin_amdgcn_rcpf(1.0f + en); // 1/(1+e), v_rcp
            // adaptive = 1 - |sigmoid(x)-0.5| = 1 - (1-e)*0.5/(1+e)
            const float adaptive = fmaf(-(1.0f - en) * 0.5f, r, 1.0f);
            weight *= adaptive;

            // bce = max(x,0) - x*s + log1p(e);  log1p(e) = -log(r)
            const float bce = mx - xi * s - __logf(r);
            // pt = exp(-bce) = exp(x*s - max(x,0)) / (1+e)
            const float pt  = __expf(fmaf(xi, s, -mx)) * r;
            const float om  = 1.0f - pt;
            const float aw  = pos > 0.0f ? 0.25f : 0.75f;    // alpha weights

            accL = fmaf(aw * om * om * bce * weight, mi, accL);
            accM += mi;
        }
    }

    // wave reduce (WMMA), then LDS across the 8 waves of this block
    const float wl = wave_sum_wmma(accL);
    const float wm = wave_sum_wmma(accM);

    __shared__ float sL[BLOCK_SZ / 32], sM[BLOCK_SZ / 32];
    const int wid  = threadIdx.x >> 5;
    const int lane = threadIdx.x & 31;
    if (lane == 0) { sL[wid] = wl; sM[wid] = wm; }
    __syncthreads();
    if (threadIdx.x == 0) {
        float bl = 0.0f, bm = 0.0f;
        #pragma unroll
        for (int i = 0; i < BLOCK_SZ / 32; ++i) { bl += sL[i]; bm += sM[i]; }
        partials[blockIdx.x]             = bl;
        partials[gridDim.x + blockIdx.x] = bm;
    }
}

__global__ void __launch_bounds__(BLOCK_SZ)
bfl_final(const float* __restrict__ partials, float* __restrict__ out,
          int nblocks) {
    float aL = 0.0f, aM = 0.0f;
    for (int i = threadIdx.x; i < nblocks; i += blockDim.x) {
        aL += partials[i];
        aM += partials[nblocks + i];
    }
    const float wl = wave_sum_wmma(aL);
    const float wm = wave_sum_wmma(aM);

    __shared__ float sL[BLOCK_SZ / 32], sM[BLOCK_SZ / 32];
    const int wid  = threadIdx.x >> 5;
    const int lane = threadIdx.x & 31;
    if (lane == 0) { sL[wid] = wl; sM[wid] = wm; }
    __syncthreads();
    if (threadIdx.x == 0) {
        float s = 0.0f, m = 0.0f;
        #pragma unroll
        for (int i = 0; i < BLOCK_SZ / 32; ++i) { s += sL[i]; m += sM[i]; }
        out[0] = (m > 0.0f) ? (s / m) : 0.0f;
    }
}

extern "C" void kernel_launch(void* const* d_in, const int* in_sizes, int n_in,
                              void* d_out, int out_size, void* d_ws, size_t ws_size,
                              hipStream_t stream) {
    const float* x  = (const float*)d_in[0];   // inputs  [B,T] f32
    const int*   tg = (const int*)d_in[1];     // targets [B,T] i32
    const float* mk = (const float*)d_in[2];   // mask    [B,T] f32
    float* out      = (float*)d_out;           // scalar f32
    float* partials = (float*)d_ws;            // 2 * N_BLOCKS floats

    const int n    = in_sizes[0];              // B*T, multiple of 4
    const int nvec = n >> 2;

    bfl_main<<<N_BLOCKS, BLOCK_SZ, 0, stream>>>(x, tg, mk, partials, nvec);
    bfl_final<<<1, BLOCK_SZ, 0, stream>>>(partials, out, N_BLOCKS);
}